// LGCNConv_59854664237752
// MI455X (gfx1250) — compile-verified
//
#include <hip/hip_runtime.h>
#include <stdint.h>

#define D       64
#define TPB     256
#define WPB     8     // waves per block (wave32)
#define CHUNK   32    // edges per wave-chunk (one index per lane)

// ---------------- CDNA5 async copy: global -> LDS (one B32 per lane) --------
__device__ __forceinline__ void async_ld_b32(uint32_t lds_addr, const void* gaddr) {
  uint64_t a = (uint64_t)(uintptr_t)gaddr;
  asm volatile("global_load_async_to_lds_b32 %0, %1, off"
               :: "v"(lds_addr), "v"(a) : "memory");
}

// ---------------- zero (float4 granularity) ---------------------------------
__global__ void zero4_kernel(float4* __restrict__ p, int n4) {
  int i = blockIdx.x * blockDim.x + threadIdx.x;
  if (i < n4) p[i] = make_float4(0.f, 0.f, 0.f, 0.f);
}

// ---------------- degree counting (integer HW atomics) ----------------------
__global__ void degree_kernel(const int* __restrict__ uidx, const int* __restrict__ sidx,
                              int* __restrict__ udeg, int* __restrict__ sdeg, int n) {
  int i = blockIdx.x * blockDim.x + threadIdx.x;
  if (i < n) {
    atomicAdd(&udeg[uidx[i]], 1);
    atomicAdd(&sdeg[sidx[i]], 1);
  }
}

// ---------------- deg -> 1/sqrt(deg) in place (isolated -> 1e-6) ------------
__global__ void invsqrt_kernel(int* deg_in, float* inv_out, int n) {
  int i = blockIdx.x * blockDim.x + threadIdx.x;
  if (i < n) {
    int c = deg_in[i];
    float d = (c == 0) ? 1e-6f : (float)c;
    inv_out[i] = rsqrtf(d);
  }
}

// Process one edge with a full wave: lanes 0..15 move the user row (float4
// each) into spot_out[s]; lanes 16..31 move the spot row into user_out[u].
// One b128 load + four f32 hardware atomics per lane.
__device__ __forceinline__ void process_edge(
    int u, int s, float wu, float ws2, int lane,
    const float* __restrict__ user_x, const float* __restrict__ spot_x,
    float* __restrict__ user_out, float* __restrict__ spot_out) {
  const int  m  = lane & 15;
  const bool lo = lane < 16;
  const float* src_row = lo ? (user_x + (size_t)u * D) : (spot_x + (size_t)s * D);
  float*       dst_row = lo ? (spot_out + (size_t)s * D) : (user_out + (size_t)u * D);
  const float  w       = lo ? wu : ws2;
  const float4 v = *(const float4*)(src_row + 4 * m);
  unsafeAtomicAdd(dst_row + 4 * m + 0, v.x * w);
  unsafeAtomicAdd(dst_row + 4 * m + 1, v.y * w);
  unsafeAtomicAdd(dst_row + 4 * m + 2, v.z * w);
  unsafeAtomicAdd(dst_row + 4 * m + 3, v.w * w);
}

// ---------------- main edge gather/scatter ----------------------------------
// Indices double-buffered into LDS via async DMA (ASYNCcnt), broadcast with
// v_readlane; rows prefetched (global_prefetch_b8); scatter via fp32 atomics.
__global__ __launch_bounds__(TPB) void edge_kernel(
    const float* __restrict__ user_x, const float* __restrict__ spot_x,
    const int* __restrict__ user_idx, const int* __restrict__ spot_idx,
    const float* __restrict__ inv_u, const float* __restrict__ inv_s,
    float* __restrict__ user_out, float* __restrict__ spot_out, int n_edges) {
  __shared__ int s_buf[WPB][2][2][CHUNK];  // [wave][parity][u/s][lane]
  const int lane  = threadIdx.x & 31;
  const int wib   = threadIdx.x >> 5;
  const int wave  = blockIdx.x * WPB + wib;
  const int nwave = gridDim.x * WPB;
  const int step  = nwave * CHUNK;

  const int base = wave * CHUNK;
  if (base >= n_edges) return;

  uint32_t lds_u[2], lds_s[2];
  lds_u[0] = (uint32_t)(uintptr_t)&s_buf[wib][0][0][lane];
  lds_s[0] = (uint32_t)(uintptr_t)&s_buf[wib][0][1][lane];
  lds_u[1] = (uint32_t)(uintptr_t)&s_buf[wib][1][0][lane];
  lds_s[1] = (uint32_t)(uintptr_t)&s_buf[wib][1][1][lane];

  {  // prologue: stage first chunk's indices
    int e = min(base + lane, n_edges - 1);
    async_ld_b32(lds_u[0], user_idx + e);
    async_ld_b32(lds_s[0], spot_idx + e);
  }

  int p = 0;
  for (int chunk = base; chunk < n_edges; chunk += step) {
    const int next = chunk + step;
    if (next < n_edges) {  // issue next chunk, then wait only for current one
      int e = min(next + lane, n_edges - 1);
      async_ld_b32(lds_u[p ^ 1], user_idx + e);
      async_ld_b32(lds_s[p ^ 1], spot_idx + e);
      asm volatile("s_wait_asynccnt 0x2" ::: "memory");
    } else {
      asm volatile("s_wait_asynccnt 0x0" ::: "memory");
    }

    const int uvals = s_buf[wib][p][0][lane];  // edge 'lane' of this chunk
    const int svals = s_buf[wib][p][1][lane];

    // warm WGP$/L2: each lane prefetches both 128B lines of its edge's rows
    const float* upre = user_x + (size_t)uvals * D;
    const float* spre = spot_x + (size_t)svals * D;
    __builtin_prefetch(upre, 0, 3);
    __builtin_prefetch(upre + 32, 0, 3);
    __builtin_prefetch(spre, 0, 3);
    __builtin_prefetch(spre + 32, 0, 3);

    // one vector gather of all 32 edges' normalization weights
    const float wu_l = inv_u[uvals];
    const float ws_l = inv_s[svals];

    const int cnt = min(CHUNK, n_edges - chunk);
    if (cnt == CHUNK) {
#pragma unroll 4
      for (int i = 0; i < CHUNK; ++i) {
        const int u   = __builtin_amdgcn_readlane(uvals, i);
        const int s   = __builtin_amdgcn_readlane(svals, i);
        const float wu  = __int_as_float(__builtin_amdgcn_readlane(__float_as_int(wu_l), i));
        const float ws2 = __int_as_float(__builtin_amdgcn_readlane(__float_as_int(ws_l), i));
        process_edge(u, s, wu, ws2, lane, user_x, spot_x, user_out, spot_out);
      }
    } else {
      for (int i = 0; i < cnt; ++i) {
        const int u   = __builtin_amdgcn_readlane(uvals, i);
        const int s   = __builtin_amdgcn_readlane(svals, i);
        const float wu  = __int_as_float(__builtin_amdgcn_readlane(__float_as_int(wu_l), i));
        const float ws2 = __int_as_float(__builtin_amdgcn_readlane(__float_as_int(ws_l), i));
        process_edge(u, s, wu, ws2, lane, user_x, spot_x, user_out, spot_out);
      }
    }
    p ^= 1;
  }
}

// ---------------- destination-side normalization ----------------------------
__global__ void scale_kernel(float4* __restrict__ out, const float* __restrict__ inv, int n4) {
  int i = blockIdx.x * blockDim.x + threadIdx.x;
  if (i < n4) {
    const float w = inv[i >> 4];  // 16 float4 per 64-float row
    float4 v = out[i];
    v.x *= w; v.y *= w; v.z *= w; v.w *= w;
    out[i] = v;
  }
}

static inline int cdiv(int a, int b) { return (a + b - 1) / b; }

extern "C" void kernel_launch(void* const* d_in, const int* in_sizes, int n_in,
                              void* d_out, int out_size, void* d_ws, size_t ws_size,
                              hipStream_t stream) {
  (void)n_in; (void)ws_size;
  const float* user_x   = (const float*)d_in[0];
  const float* spot_x   = (const float*)d_in[1];
  const int*   user_idx = (const int*)d_in[2];
  const int*   spot_idx = (const int*)d_in[3];
  const int n_users = in_sizes[0] / D;   // 200000
  const int n_spots = in_sizes[1] / D;   // 50000
  const int n_edges = in_sizes[2];       // 3200000

  float* user_out = (float*)d_out;
  float* spot_out = user_out + (size_t)n_users * D;

  int* udeg = (int*)d_ws;                // n_users ints, becomes inv_sqrt_user
  int* sdeg = udeg + n_users;            // n_spots ints, becomes inv_sqrt_spot
  float* inv_u = (float*)udeg;
  float* inv_s = (float*)sdeg;
  const int tot = n_users + n_spots;     // 250000 (multiple of 4)

  // zero accumulators (ws degrees + full output) every call: deterministic
  zero4_kernel<<<cdiv(tot / 4, TPB), TPB, 0, stream>>>((float4*)d_ws, tot / 4);
  zero4_kernel<<<cdiv(out_size / 4, TPB), TPB, 0, stream>>>((float4*)d_out, out_size / 4);

  degree_kernel<<<cdiv(n_edges, TPB), TPB, 0, stream>>>(user_idx, spot_idx, udeg, sdeg, n_edges);
  invsqrt_kernel<<<cdiv(tot, TPB), TPB, 0, stream>>>((int*)d_ws, (float*)d_ws, tot);

  int chunks = cdiv(n_edges, WPB * CHUNK);
  int blocks = chunks < 3200 ? chunks : 3200;   // ~25.6k waves, ~4 chunks each
  edge_kernel<<<blocks, TPB, 0, stream>>>(user_x, spot_x, user_idx, spot_idx,
                                          inv_u, inv_s, user_out, spot_out, n_edges);

  scale_kernel<<<cdiv(n_users * 16, TPB), TPB, 0, stream>>>((float4*)user_out, inv_u, n_users * 16);
  scale_kernel<<<cdiv(n_spots * 16, TPB), TPB, 0, stream>>>((float4*)spot_out, inv_s, n_spots * 16);
}